// ExponentialUnitNorm_13589276524636
// MI455X (gfx1250) — compile-verified
//
#include <hip/hip_runtime.h>

// ---------------------------------------------------------------------------
// ExponentialUnitNorm for MI455X (gfx1250)
// EMA scan recast as 16x16x16 GEMM chains on v_wmma_f32_16x16x4_f32,
// async global->LDS staging, wave-private double buffering (no barriers).
// ---------------------------------------------------------------------------

typedef __attribute__((ext_vector_type(2))) float v2f;
typedef __attribute__((ext_vector_type(8))) float v8f;
typedef __attribute__((ext_vector_type(4))) int   v4i;

typedef __attribute__((address_space(1))) v4i g_v4i;  // global <4 x i32>
typedef __attribute__((address_space(3))) v4i l_v4i;  // LDS    <4 x i32>

#ifndef __has_builtin
#define __has_builtin(x) 0
#endif

#if __has_builtin(__builtin_amdgcn_global_load_async_to_lds_b128)
#define HAS_ASYNC_LDS 1
#else
#define HAS_ASYNC_LDS 0
#endif
#if __has_builtin(__builtin_amdgcn_s_wait_asynccnt)
#define HAS_WAIT_ASYNC 1
#else
#define HAS_WAIT_ASYNC 0
#endif

constexpr int Bn = 32, Fn = 513, Tn = 4000;
constexpr int NROWG  = (Bn * Fn) / 32;   // 513 groups of 32 rows
constexpr int NCHUNK = 2;                // T split with EMA warm-up
constexpr int NWAVE  = NROWG * NCHUNK;   // 1026 independent waves
constexpr int WPB    = 4;                // waves per block (128 threads)
constexpr int TT     = 32;               // time tile (2 WMMA chains of 16)
constexpr int SIN    = 36;               // padded LDS row stride (floats)
constexpr int LDSW   = 3 * 32 * SIN;     // floats per wave: in0,in1,out tiles
constexpr float ALPHA = 0.969f;          // round(exp(-0.032), 3) as in reference

__device__ __forceinline__ float powA(int e) {
  float r = 1.0f, b = ALPHA;
  while (e) { if (e & 1) r *= b; b *= b; e >>= 1; }
  return r;
}

template <int N> __device__ __forceinline__ void wait_async() {
#if HAS_ASYNC_LDS && HAS_WAIT_ASYNC
  __builtin_amdgcn_s_wait_asynccnt(N);
#elif HAS_ASYNC_LDS
  asm volatile("s_wait_asynccnt %0" :: "n"(N) : "memory");
#endif
  asm volatile("" ::: "memory");   // keep LDS reads after the wait
}

// Stage one 32-row x 32-timestep tile into LDS (coalesced 512B/instr).
__device__ __forceinline__ void load_tile(const float* __restrict__ xg,
                                          float* inbuf, int rg, int tglob,
                                          int lane) {
  const int lr = lane >> 3;          // 4 rows per instruction
  const int lc = (lane & 7) << 2;    // 4 floats per lane
#pragma unroll
  for (int jr = 0; jr < 8; ++jr) {
    const float* gp = xg + ((size_t)(rg * 32 + jr * 4 + lr) * Tn + (tglob + lc));
    float*       lp = inbuf + ((jr * 4 + lr) * SIN + lc);
#if HAS_ASYNC_LDS
    __builtin_amdgcn_global_load_async_to_lds_b128((g_v4i*)gp, (l_v4i*)lp, 0, 0);
#else
    *(float4*)lp = *(const float4*)gp;  // fallback: sync copy via VGPRs
#endif
  }
}

__global__ void __launch_bounds__(WPB * 32)
ema_unit_norm(const float* __restrict__ x, float* __restrict__ out) {
  __shared__ float lds[WPB * LDSW];

  const int tid  = threadIdx.x;
  const int wv   = tid >> 5;
  const int lane = tid & 31;
  const int w    = blockIdx.x * WPB + wv;   // wave-uniform
  if (w >= NWAVE) return;

  const int chunk = w & 1;
  const int rg    = w >> 1;                 // row-group [0, 513)

  float* inb0 = &lds[wv * LDSW];
  float* inb1 = inb0 + 32 * SIN;
  float* outb = inb1 + 32 * SIN;

  const int hh  = lane >> 4;   // half-wave id (WMMA lane split)
  const int lq  = lane & 15;
  const int off = hh * 8;      // C/D: VGPR j holds M = j + off

  // Constant A operand: M[t][k] = (1-a)*a^(t-k), lower triangular.
  // 32-bit A 16x4 layout: VGPR0 = {K=0 | K=2}, VGPR1 = {K=1 | K=3}, rows in lq.
  v2f Am[4];
#pragma unroll
  for (int i = 0; i < 4; ++i) {
    const int kx = 4 * i + hh * 2;
    Am[i].x = (kx     <= lq) ? (1.0f - ALPHA) * powA(lq - kx)     : 0.0f;
    Am[i].y = (kx + 1 <= lq) ? (1.0f - ALPHA) * powA(lq - kx - 1) : 0.0f;
  }
  // Carry weights: C[t][n] = a^(t+1) * s_prev[n], t = j + off.
  float apw[8];
#pragma unroll
  for (int j = 0; j < 8; ++j) apw[j] = powA(j + off + 1);

  // Initial states: linspace(0.001, 0.0001, 513) per frequency.
  const int r0 = rg * 32 + lq;
  const int r1 = r0 + 16;
  float sp2[2];
  sp2[0] = 0.001f - 0.0009f * (float)(r0 % Fn) * (1.0f / 512.0f);
  sp2[1] = 0.001f - 0.0009f * (float)(r1 % Fn) * (1.0f / 512.0f);

  // Chunk 0: t in [0,2016) fully written. Chunk 1: warm-up [1504,2016)
  // (alpha^512 ~ 1e-7 => init truncation below f32 noise), write [2016,4000).
  const int t_begin    = chunk ? 1504 : 0;
  const int n_tiles    = chunk ? 78   : 63;
  const int warm_tiles = chunk ? 16   : 0;

  load_tile(x, inb0, rg, t_begin, lane);

  for (int tile = 0; tile < n_tiles; ++tile) {
    float* cur = (tile & 1) ? inb1 : inb0;
    float* nxt = (tile & 1) ? inb0 : inb1;
    if (tile + 1 < n_tiles) {
      load_tile(x, nxt, rg, t_begin + (tile + 1) * TT, lane);
      wait_async<8>();   // 8 in flight for next tile; current tile complete
    } else {
      wait_async<0>();
    }

    const bool dowrite = (tile >= warm_tiles);

#pragma unroll
    for (int g = 0; g < 2; ++g) {          // two independent 16-row chains
      const int go = g * 16;
      const float* rowp = cur + (go + lq) * SIN;
#pragma unroll
      for (int tt = 0; tt < TT; tt += 16) {
        v8f acc;
#pragma unroll
        for (int j = 0; j < 8; ++j) acc[j] = apw[j] * sp2[g];  // carry -> C
#pragma unroll
        for (int i = 0; i < 4; ++i) {
          // B 4x16 layout: VGPR0 = {K=0 | K=2}, VGPR1 = {K=1 | K=3}, cols in lq
          const int kk = tt + 4 * i + hh * 2;
          const float x0 = rowp[kk];
          const float x1 = rowp[kk + 1];
          v2f b;
          b.x = __builtin_amdgcn_sqrtf(fmaxf(x0, 1e-10f));
          b.y = __builtin_amdgcn_sqrtf(fmaxf(x1, 1e-10f));
          acc = __builtin_amdgcn_wmma_f32_16x16x4_f32(
              false, Am[i], false, b, (short)0, acc, false, false);
        }
        // New carry = D[15][n], held in acc[7] of lane 16+n.
        sp2[g] = __shfl(acc[7], 16 + lq, 32);

        if (dowrite) {
          const float4 xa = *(const float4*)(rowp + tt + off);
          const float4 xb = *(const float4*)(rowp + tt + off + 4);
          float4 oa, ob;
          oa.x = xa.x * __builtin_amdgcn_rsqf(acc[0]);
          oa.y = xa.y * __builtin_amdgcn_rsqf(acc[1]);
          oa.z = xa.z * __builtin_amdgcn_rsqf(acc[2]);
          oa.w = xa.w * __builtin_amdgcn_rsqf(acc[3]);
          ob.x = xb.x * __builtin_amdgcn_rsqf(acc[4]);
          ob.y = xb.y * __builtin_amdgcn_rsqf(acc[5]);
          ob.z = xb.z * __builtin_amdgcn_rsqf(acc[6]);
          ob.w = xb.w * __builtin_amdgcn_rsqf(acc[7]);
          float* op = outb + (go + lq) * SIN + tt + off;
          *(float4*)op       = oa;
          *(float4*)(op + 4) = ob;
        }
      }
    }

    if (dowrite) {  // coalesced flush: 4 rows x 128B per instruction
      const int tg = t_begin + tile * TT;
      const int fr = lane >> 3;
      const int fc = (lane & 7) << 2;
#pragma unroll
      for (int jr = 0; jr < 8; ++jr) {
        const int r = jr * 4 + fr;
        const float4 v = *(const float4*)&outb[r * SIN + fc];
        *(float4*)(out + (size_t)(rg * 32 + r) * Tn + tg + fc) = v;
      }
    }
  }
}

extern "C" void kernel_launch(void* const* d_in, const int* in_sizes, int n_in,
                              void* d_out, int out_size, void* d_ws, size_t ws_size,
                              hipStream_t stream) {
  (void)in_sizes; (void)n_in; (void)d_ws; (void)ws_size; (void)out_size;
  const float* x = (const float*)d_in[0];
  float* out = (float*)d_out;
  const dim3 grid((NWAVE + WPB - 1) / WPB);  // 257 blocks
  const dim3 block(WPB * 32);                // 128 threads = 4 waves
  hipLaunchKernelGGL(ema_unit_norm, grid, block, 0, stream, x, out);
}